// GraphAttentionNetwork_27986006901171
// MI455X (gfx1250) — compile-verified
//
#include <hip/hip_runtime.h>

// GAT forward for B=4, D=128, K=512, f32 end-to-end.
// K1: left/right = xt@W1(+b) / xt@W2      via v_wmma_f32_16x16x4_f32
// K2: e[b,i,j] = sum_d lrelu(l+r)*a + bias; right tile via TDM (padded LDS),
//     left chunks via double-buffered GLOBAL_LOAD_ASYNC_TO_LDS_B128
// K3: row softmax
// K4: agg = attn@xt, out = sigmoid(agg)+xt via v_wmma_f32_16x16x4_f32 (2 M-tiles/wave)

#define BB 4
#define DD 128
#define KK 512
#define GAT_ALPHA 0.2f

typedef __attribute__((ext_vector_type(2))) float        v2f;
typedef __attribute__((ext_vector_type(8))) float        v8f;
typedef __attribute__((ext_vector_type(4))) int          v4i;
typedef __attribute__((ext_vector_type(8))) int          v8i;
typedef __attribute__((ext_vector_type(4))) unsigned int v4u;

#if defined(__gfx1250__) && __has_builtin(__builtin_amdgcn_global_load_async_to_lds_b128) && __has_builtin(__builtin_amdgcn_s_wait_asynccnt)
#define USE_ASYNC_LDS 1
#else
#define USE_ASYNC_LDS 0
#endif

#if USE_ASYNC_LDS && __has_builtin(__builtin_amdgcn_tensor_load_to_lds) && __has_builtin(__builtin_amdgcn_s_wait_tensorcnt)
#define USE_TDM 1
#else
#define USE_TDM 0
#endif

#if defined(__gfx1250__)
typedef v4i __attribute__((address_space(1)))* as1_v4i_ptr;
typedef v4i __attribute__((address_space(3)))* as3_v4i_ptr;
typedef void __attribute__((address_space(3)))* as3_void_ptr;
#endif

// ---------------------------------------------------------------------------
// K1: left = xt@W1 + lin_b ; right = xt@W2
// One wave per 16x16 output tile, computing BOTH left and right (A reuse).
// grid: 128 blocks (M-tiles), 8 waves/block (N-tiles). K-loop: 32 wmma steps x2.
// xt[r,d] = x[b, d, i] (r = b*512+i), W1=lin_w[0:128,:], W2=lin_w[128:256,:].
// ---------------------------------------------------------------------------
__global__ __launch_bounds__(256) void gat_lin_kernel(
    const float* __restrict__ x, const float* __restrict__ lin_w,
    const float* __restrict__ lin_b, float* __restrict__ left,
    float* __restrict__ right)
{
  const int lane = threadIdx.x & 31;
  const int wave = threadIdx.x >> 5;
  const int lm = lane & 15;       // row/col-in-tile
  const int lg = lane >> 4;       // half-wave group
  const int m0 = blockIdx.x * 16; // output row tile (flattened b*K+i)
  const int n0 = wave * 16;       // output col tile (feature)
  const int b  = m0 / KK;
  const int i  = (m0 % KK) + lm;  // K-dim column inside x
  const int n  = n0 + lm;

  const float* xb = x + b * (DD * KK);

  const float bias = lin_b[n];
  v8f accL, accR;
  #pragma unroll
  for (int v = 0; v < 8; ++v) { accL[v] = bias; accR[v] = 0.f; }

  for (int k0 = 0; k0 < DD; k0 += 4) {
    const int kk = k0 + 2 * lg;   // A: lanes 0-15 carry K=k0,k0+1 ; 16-31 carry K=k0+2,k0+3
    v2f A, B1, B2;
    A.x  = xb[kk * KK + i];
    A.y  = xb[(kk + 1) * KK + i];
    B1.x = lin_w[kk * DD + n];
    B1.y = lin_w[(kk + 1) * DD + n];
    B2.x = lin_w[(kk + DD) * DD + n];
    B2.y = lin_w[(kk + DD + 1) * DD + n];
    __builtin_prefetch(xb + (kk + 4) * KK + i, 0, 3);
    accL = __builtin_amdgcn_wmma_f32_16x16x4_f32(false, A, false, B1, (short)0, accL, false, false);
    accR = __builtin_amdgcn_wmma_f32_16x16x4_f32(false, A, false, B2, (short)0, accR, false, false);
  }

  // C/D layout: VGPR v -> M = m0+v (lanes 0-15), m0+8+v (lanes 16-31); N = n0+lm
  const int mbase = m0 + 8 * lg;
  #pragma unroll
  for (int v = 0; v < 8; ++v) {
    const int m = mbase + v;
    left [m * DD + n] = accL[v];
    right[m * DD + n] = accR[v];
  }
}

// ---------------------------------------------------------------------------
// K2: scores. Block = (b, 64-wide j tile).
// right tile -> LDS via TDM with hardware padding: 1 DWORD pad per 128 DWORDs
// reproduces the 129-float row stride (bank = (j + d) % 64, conflict-free).
// left chunks -> double-buffered async-to-LDS pipeline (copy/compute overlap).
// ---------------------------------------------------------------------------
__global__ __launch_bounds__(256) void gat_scores_kernel(
    const float* __restrict__ left, const float* __restrict__ right,
    const float* __restrict__ avec, const float* __restrict__ bias_mat,
    float* __restrict__ attn)
{
  __shared__ float rt[64 * 129];       // padded right tile (33 KB)
  __shared__ float lc[2][32 * DD];     // double-buffered left chunks (32 KB)
  __shared__ float sa[DD];

  const int tid  = threadIdx.x;
  const int wave = tid >> 5;
  const int jt   = blockIdx.x;         // 0..7
  const int b    = blockIdx.y;         // 0..3

  if (tid < DD) sa[tid] = avec[tid];

  const float* rbase = right + (size_t)(b * KK + jt * 64) * DD;
  const float* lbase = left + (size_t)b * KK * DD;

  // ---- right tile fill ----
#if USE_TDM
  if (wave == 0) {
    const unsigned ldsa = (unsigned)(unsigned long long)(as3_void_ptr)&rt[0];
    const unsigned long long ga = (unsigned long long)rbase;
    v4u g0;
    g0.x = 1u;                                        // count=1, user descriptor
    g0.y = ldsa;                                      // lds_addr (bytes)
    g0.z = (unsigned)(ga & 0xFFFFFFFFu);              // global_addr[31:0]
    g0.w = (unsigned)((ga >> 32) & 0x01FFFFFFu)       // global_addr[56:32]
         | (2u << 30);                                // type = 2 ("image")
    v8i g1;
    g1[0] = (int)((2u << 16)   // data_size = 4 bytes
                | (1u << 20)   // pad_enable
                | (6u << 22)   // pad_interval: 128 DWORDs
                | (0u << 25)); // pad_amount: 1 DWORD  -> LDS row stride 129
    g1[1] = (int)(128u << 16); // tensor_dim0 = 128 (row length, elements)
    g1[2] = (int)(64u << 16);  // tensor_dim1 = 64 (rows in this tile)
    g1[3] = (int)(128u << 16); // tile_dim0 = 128
    g1[4] = (int)(64u);        // tile_dim1 = 64, tile_dim2 = 0
    g1[5] = (int)(128u);       // tensor_dim0_stride = 128 elements
    g1[6] = 0;
    g1[7] = 0;
    v4i gz4 = {0, 0, 0, 0};
    v8i gz8 = {0, 0, 0, 0, 0, 0, 0, 0};
    __builtin_amdgcn_tensor_load_to_lds(g0, g1, gz4, gz4, gz8, 0);
    __builtin_amdgcn_s_wait_tensorcnt(0);
  }
#elif USE_ASYNC_LDS
  for (int t = 0; t < 8; ++t) {
    const int q = tid + 256 * t;       // float4 index, 0..2047
    const int j = q >> 5;
    const int d = (q & 31) << 2;
    __builtin_amdgcn_global_load_async_to_lds_b128(
        (as1_v4i_ptr)(rbase + (q << 2)),
        (as3_v4i_ptr)&rt[j * 129 + d], 0, 0);
  }
  __builtin_amdgcn_s_wait_asynccnt(0);
#else
  for (int r = tid; r < 64 * DD; r += 256)
    rt[(r >> 7) * 129 + (r & 127)] = rbase[r];
#endif

  // ---- left chunk pipeline helpers ----
#if USE_ASYNC_LDS
#define GAT_ISSUE_LC(dst, ic)                                                  \
  do {                                                                         \
    for (int t_ = 0; t_ < 4; ++t_) {                                           \
      const int q_ = tid + 256 * t_; /* float4 index, 0..1023 */               \
      __builtin_amdgcn_global_load_async_to_lds_b128(                          \
          (as1_v4i_ptr)(lbase + (size_t)(ic) * (32 * DD) + (q_ << 2)),         \
          (as3_v4i_ptr)&(dst)[q_ << 2], 0, 0);                                 \
    }                                                                          \
  } while (0)
#define GAT_WAIT_LC() __builtin_amdgcn_s_wait_asynccnt(0)
#else
#define GAT_ISSUE_LC(dst, ic)                                                  \
  do {                                                                         \
    for (int r_ = tid; r_ < 32 * DD; r_ += 256)                                \
      (dst)[r_] = lbase[(size_t)(ic) * (32 * DD) + r_];                        \
  } while (0)
#define GAT_WAIT_LC() ((void)0)
#endif

  GAT_ISSUE_LC(lc[0], 0);

  const int jl   = tid & 63;
  const int isub = tid >> 6;           // uniform within each wave32
  const int jg   = jt * 64 + jl;
  const float* rrow = &rt[jl * 129];

  for (int ic = 0; ic < 16; ++ic) {    // 16 chunks of 32 i-rows
    const float* cur = lc[ic & 1];
    GAT_WAIT_LC();
    __syncthreads();                   // cur chunk resident; prev compute done
    if (ic + 1 < 16) GAT_ISSUE_LC(lc[(ic + 1) & 1], ic + 1);

    #pragma unroll 1
    for (int s = 0; s < 8; ++s) {
      const int il = s * 4 + isub;
      const int ig = ic * 32 + il;
      const float* lrow = &cur[il * DD];  // wave-uniform -> LDS broadcast reads
      float acc = 0.f;
      #pragma unroll 4
      for (int d = 0; d < DD; ++d) {
        const float p = lrow[d] + rrow[d];
        const float h = p > 0.f ? p : GAT_ALPHA * p;
        acc = fmaf(h, sa[d], acc);
      }
      attn[((size_t)(b * KK + ig)) * KK + jg] = acc + bias_mat[ig * KK + jg];
    }
  }
#undef GAT_ISSUE_LC
#undef GAT_WAIT_LC
}

// ---------------------------------------------------------------------------
// K3: softmax over j (row length 512), one block per row, 2 elems/thread.
// ---------------------------------------------------------------------------
__global__ __launch_bounds__(256) void gat_softmax_kernel(float* __restrict__ attn)
{
  __shared__ float red[256];
  const int t = threadIdx.x;
  float* row = attn + (size_t)blockIdx.x * KK;
  const float e0 = row[t], e1 = row[t + 256];

  red[t] = fmaxf(e0, e1);
  __syncthreads();
  for (int off = 128; off > 0; off >>= 1) {
    if (t < off) red[t] = fmaxf(red[t], red[t + off]);
    __syncthreads();
  }
  const float M = red[0];
  __syncthreads();

  const float x0 = __expf(e0 - M), x1 = __expf(e1 - M);
  red[t] = x0 + x1;
  __syncthreads();
  for (int off = 128; off > 0; off >>= 1) {
    if (t < off) red[t] += red[t + off];
    __syncthreads();
  }
  const float inv = 1.f / red[0];
  row[t]       = x0 * inv;
  row[t + 256] = x1 * inv;
}

// ---------------------------------------------------------------------------
// K4: agg = attn @ xt per batch (512x512x128), sigmoid + residual, store
// transposed as (B, D, K). Each wave: 32(i) x 16(d) -> two WMMA tiles share
// the (strided) B fragment; 128 k-steps x 2 wmma.
// ---------------------------------------------------------------------------
__global__ __launch_bounds__(256) void gat_agg_kernel(
    const float* __restrict__ attn, const float* __restrict__ x,
    float* __restrict__ out)
{
  const int lane = threadIdx.x & 31;
  const int wave = threadIdx.x >> 5;
  const int lm = lane & 15;
  const int lg = lane >> 4;
  const int b  = blockIdx.y;
  const int m0 = blockIdx.x * 32;  // i super-tile (two 16-row tiles)
  const int n0 = wave * 16;        // d tile
  const int n  = n0 + lm;

  const float* arow0 = attn + ((size_t)(b * KK + m0 + lm)) * KK;
  const float* arow1 = arow0 + (size_t)16 * KK;
  const float* xb    = x + b * (DD * KK);
  const float* bcol  = xb + n * KK;

  v8f acc0, acc1;
  #pragma unroll
  for (int v = 0; v < 8; ++v) { acc0[v] = 0.f; acc1[v] = 0.f; }

  for (int k0 = 0; k0 < KK; k0 += 4) {
    const int kk = k0 + 2 * lg;
    v2f A0, A1, Bv;
    A0.x = arow0[kk]; A0.y = arow0[kk + 1];
    A1.x = arow1[kk]; A1.y = arow1[kk + 1];
    Bv.x = bcol[kk];  Bv.y = bcol[kk + 1];
    acc0 = __builtin_amdgcn_wmma_f32_16x16x4_f32(false, A0, false, Bv, (short)0, acc0, false, false);
    acc1 = __builtin_amdgcn_wmma_f32_16x16x4_f32(false, A1, false, Bv, (short)0, acc1, false, false);
  }

  #pragma unroll
  for (int v = 0; v < 8; ++v) {
    const int mm0 = m0 + 8 * lg + v;
    const int mm1 = mm0 + 16;
    const float s0 = 1.f / (1.f + __expf(-acc0[v]));
    const float s1 = 1.f / (1.f + __expf(-acc1[v]));
    out[(size_t)b * (DD * KK) + n * KK + mm0] = s0 + xb[n * KK + mm0];
    out[(size_t)b * (DD * KK) + n * KK + mm1] = s1 + xb[n * KK + mm1];
  }
}

// ---------------------------------------------------------------------------
extern "C" void kernel_launch(void* const* d_in, const int* in_sizes, int n_in,
                              void* d_out, int out_size, void* d_ws, size_t ws_size,
                              hipStream_t stream)
{
  (void)in_sizes; (void)n_in; (void)out_size; (void)ws_size;

  const float* x      = (const float*)d_in[0];  // (B, D, K)
  const float* lin_w  = (const float*)d_in[1];  // (2D, D)
  const float* lin_b  = (const float*)d_in[2];  // (D,)
  const float* avec   = (const float*)d_in[3];  // (D,)
  const float* bias   = (const float*)d_in[4];  // (K, K)
  float* out = (float*)d_out;                   // (B, D, K)

  float* wsf   = (float*)d_ws;
  float* left  = wsf;                              // B*K*D
  float* right = wsf + (size_t)BB * KK * DD;       // B*K*D
  float* attn  = wsf + (size_t)2 * BB * KK * DD;   // B*K*K

  gat_lin_kernel    <<<dim3((BB * KK) / 16), dim3(256), 0, stream>>>(x, lin_w, lin_b, left, right);
  gat_scores_kernel <<<dim3(KK / 64, BB),    dim3(256), 0, stream>>>(left, right, avec, bias, attn);
  gat_softmax_kernel<<<dim3(BB * KK),        dim3(256), 0, stream>>>(attn);
  gat_agg_kernel    <<<dim3(KK / 32, BB),    dim3(256), 0, stream>>>(attn, x, out);
}